// HRAPilLayer_56538949484840
// MI455X (gfx1250) — compile-verified
//
#include <hip/hip_runtime.h>
#include <hip/hip_bf16.h>
#include <math.h>

typedef __attribute__((ext_vector_type(2))) float v2f;
typedef __attribute__((ext_vector_type(8))) float v8f;

#define N_ROWS   4096
#define D_IN     2048
#define D_HID    8192
#define D_OUT    2048
#define RANK     8

// ---------------------------------------------------------------------------
// Block-wide sum reduction (256 threads, wave32): shfl within wave, LDS across.
// ---------------------------------------------------------------------------
__device__ __forceinline__ float block_reduce_sum_256(float p, float* redw) {
    #pragma unroll
    for (int off = 16; off > 0; off >>= 1) p += __shfl_down(p, off, 32);
    int t = threadIdx.x;
    __syncthreads();                   // protect redw from previous round
    if ((t & 31) == 0) redw[t >> 5] = p;
    __syncthreads();
    float s = 0.f;
    #pragma unroll
    for (int w = 0; w < 8; ++w) s += redw[w];
    return s;
}

// ---------------------------------------------------------------------------
// Kernel 1: modified Gram-Schmidt on hra_u (2048 x 8), single block of 256.
// Columns live in registers: u[j][e] holds element d = e*256 + tid of col j.
// Writes U column-major: U[j*2048 + d].
// ---------------------------------------------------------------------------
__global__ __launch_bounds__(256)
void hra_gs_kernel(const float* __restrict__ hra_u, float* __restrict__ U) {
    __shared__ float redw[8];
    const int t = threadIdx.x;
    float u[RANK][8];
    #pragma unroll
    for (int e = 0; e < 8; ++e) {
        int d = e * 256 + t;
        #pragma unroll
        for (int j = 0; j < RANK; ++j) u[j][e] = hra_u[d * RANK + j];
    }
    for (int i = 0; i < RANK; ++i) {
        for (int j = 0; j < i; ++j) {
            float p = 0.f;
            #pragma unroll
            for (int e = 0; e < 8; ++e) p += u[j][e] * u[i][e];
            float dot = block_reduce_sum_256(p, redw);
            #pragma unroll
            for (int e = 0; e < 8; ++e) u[i][e] -= dot * u[j][e];
        }
        float p = 0.f;
        #pragma unroll
        for (int e = 0; e < 8; ++e) p += u[i][e] * u[i][e];
        float nrm2 = block_reduce_sum_256(p, redw);
        float inv = 1.0f / (sqrtf(nrm2) + 1e-6f);
        #pragma unroll
        for (int e = 0; e < 8; ++e) u[i][e] *= inv;
    }
    #pragma unroll
    for (int j = 0; j < RANK; ++j)
        #pragma unroll
        for (int e = 0; e < 8; ++e)
            U[j * D_IN + e * 256 + t] = u[j][e];
}

// ---------------------------------------------------------------------------
// Kernel 2: y = x - 2 * (x U) U^T    (Q = I - 2 U U^T for orthonormal U)
// One 256-thread block per row of x.
// ---------------------------------------------------------------------------
__global__ __launch_bounds__(256)
void hra_apply_kernel(const float* __restrict__ x, const float* __restrict__ U,
                      float* __restrict__ y) {
    __shared__ float red[8][RANK];
    const int t   = threadIdx.x;
    const size_t row = blockIdx.x;
    const float* xr = x + row * D_IN;

    float xv[8];
    #pragma unroll
    for (int e = 0; e < 8; ++e) xv[e] = xr[e * 256 + t];

    float c[RANK];
    #pragma unroll
    for (int j = 0; j < RANK; ++j) {
        float p = 0.f;
        #pragma unroll
        for (int e = 0; e < 8; ++e) p += xv[e] * U[j * D_IN + e * 256 + t];
        #pragma unroll
        for (int off = 16; off > 0; off >>= 1) p += __shfl_down(p, off, 32);
        if ((t & 31) == 0) red[t >> 5][j] = p;
    }
    __syncthreads();
    #pragma unroll
    for (int j = 0; j < RANK; ++j) {
        float s = 0.f;
        #pragma unroll
        for (int w = 0; w < 8; ++w) s += red[w][j];
        c[j] = 2.0f * s;
    }
    float* yr = y + row * D_IN;
    #pragma unroll
    for (int e = 0; e < 8; ++e) {
        int d = e * 256 + t;
        float acc = xv[e];
        #pragma unroll
        for (int j = 0; j < RANK; ++j) acc -= c[j] * U[j * D_IN + d];
        yr[d] = acc;
    }
}

// ---------------------------------------------------------------------------
// Kernel 3/4: fp32 WMMA GEMM, C = epilogue(A @ B)
//   EPI == 0 : exact GELU   (h = gelu(y @ W_random))
//   EPI == 1 : + bias       (out = h @ W_out + bias)
// Block tile 128x128, K-chunk 16, double-buffered LDS (one barrier/iter).
// 8 waves, each owns a 32x64 tile (2x4 accumulators of v_wmma_f32_16x16x4_f32).
// A staged as As[m][k] (pad 18), B staged TRANSPOSED as Bt[n][k] (pad 18) so
// both A and B fragments are single 8B-aligned ds_load_b64 per tile.
// ---------------------------------------------------------------------------
#define BM 128
#define BN 128
#define BK 16
#define LDAB 18   // padded LDS k-stride (even -> 8B-aligned ds_load_b64)

template <int EPI>
__global__ __launch_bounds__(256)
void wmma_gemm_f32_kernel(const float* __restrict__ A, const float* __restrict__ B,
                          const float* __restrict__ bias, float* __restrict__ C,
                          int M, int N, int K) {
    __shared__ float As [2][BM * LDAB];   // 2 x 9216 B
    __shared__ float Bts[2][BN * LDAB];   // 2 x 9216 B (transposed: [n][k])

    const int t    = threadIdx.x;
    const int wave = t >> 5;         // 0..7
    const int lane = t & 31;
    const int wm   = wave & 3;       // 4 waves along M
    const int wn   = wave >> 2;      // 2 waves along N
    const int l16  = lane & 15;
    const int lh   = lane >> 4;      // lane half: 0/1

    const int m_blk = blockIdx.y * BM;
    const int n_blk = blockIdx.x * BN;

    v8f acc[2][4] = {};
    float4 ra[2], rb[2];

    // -------- global tile loads into registers --------
    auto load_tiles = [&](int k0) {
        #pragma unroll
        for (int i = 0; i < 2; ++i) {
            int idx = t + i * 256;                    // A: float4 unit 0..511
            ra[i] = *(const float4*)(A + (size_t)(m_blk + (idx >> 2)) * K +
                                     k0 + ((idx & 3) << 2));
        }
        #pragma unroll
        for (int i = 0; i < 2; ++i) {
            int idx = t + i * 256;                    // B: row k, 4 cols
            rb[i] = *(const float4*)(B + (size_t)(k0 + (idx >> 5)) * N +
                                     n_blk + ((idx & 31) << 2));
        }
    };
    // -------- register -> LDS stores (B transposed) --------
    auto store_tiles = [&](int buf) {
        #pragma unroll
        for (int i = 0; i < 2; ++i) {
            int idx = t + i * 256;
            float* p = &As[buf][(idx >> 2) * LDAB + ((idx & 3) << 2)];
            p[0] = ra[i].x; p[1] = ra[i].y; p[2] = ra[i].z; p[3] = ra[i].w;
        }
        #pragma unroll
        for (int i = 0; i < 2; ++i) {
            int idx = t + i * 256;
            int r   = idx >> 5;                       // k
            int c4  = (idx & 31) << 2;                // n base
            Bts[buf][(c4 + 0) * LDAB + r] = rb[i].x;
            Bts[buf][(c4 + 1) * LDAB + r] = rb[i].y;
            Bts[buf][(c4 + 2) * LDAB + r] = rb[i].z;
            Bts[buf][(c4 + 3) * LDAB + r] = rb[i].w;
        }
    };
    // -------- 4 WMMA k-steps over one staged K-chunk --------
    auto compute = [&](int buf) {
        #pragma unroll
        for (int kk = 0; kk < BK; kk += 4) {
            const int kfrag = kk + 2 * lh;            // this lane's K pair base
            v2f af[2];
            #pragma unroll
            for (int i = 0; i < 2; ++i)
                af[i] = *(const v2f*)(&As[buf][(wm * 32 + i * 16 + l16) * LDAB + kfrag]);
            v2f bf[4];
            #pragma unroll
            for (int j = 0; j < 4; ++j)
                bf[j] = *(const v2f*)(&Bts[buf][(wn * 64 + j * 16 + l16) * LDAB + kfrag]);
            #pragma unroll
            for (int i = 0; i < 2; ++i)
                #pragma unroll
                for (int j = 0; j < 4; ++j)
                    acc[i][j] = __builtin_amdgcn_wmma_f32_16x16x4_f32(
                        false, af[i], false, bf[j], (short)0, acc[i][j],
                        false, false);
        }
    };

    // -------- software pipeline: one barrier per K-chunk --------
    load_tiles(0);
    store_tiles(0);
    __syncthreads();
    int buf = 0;
    for (int k0 = BK; k0 < K; k0 += BK) {
        load_tiles(k0);          // next stage -> regs (overlaps with WMMAs)
        compute(buf);            // reads buf
        store_tiles(buf ^ 1);    // buf^1's readers finished before last barrier
        __syncthreads();
        buf ^= 1;
    }
    compute(buf);

    // -------- epilogue: D layout vgpr r -> row r + 8*lh, col l16 --------
    #pragma unroll
    for (int i = 0; i < 2; ++i) {
        #pragma unroll
        for (int j = 0; j < 4; ++j) {
            int col = n_blk + wn * 64 + j * 16 + l16;
            float bv = (EPI == 1) ? bias[col] : 0.f;
            #pragma unroll
            for (int r = 0; r < 8; ++r) {
                int row = m_blk + wm * 32 + i * 16 + r + 8 * lh;
                float v = acc[i][j][r];
                if (EPI == 0) {
                    v = 0.5f * v * (1.0f + erff(v * 0.70710678118654752f));
                } else {
                    v += bv;
                }
                C[(size_t)row * N + col] = v;
            }
        }
    }
}

// ---------------------------------------------------------------------------
extern "C" void kernel_launch(void* const* d_in, const int* in_sizes, int n_in,
                              void* d_out, int out_size, void* d_ws, size_t ws_size,
                              hipStream_t stream) {
    const float* x    = (const float*)d_in[0];   // 4096 x 2048
    const float* hra  = (const float*)d_in[1];   // 2048 x 8
    const float* Wr   = (const float*)d_in[2];   // 2048 x 8192
    const float* Wo   = (const float*)d_in[3];   // 8192 x 2048
    const float* bias = (const float*)d_in[4];   // 2048
    float* out = (float*)d_out;                  // 4096 x 2048

    char* ws = (char*)d_ws;
    float* U = (float*)ws;                                      // 64 KB
    float* y = (float*)(ws + (64u << 10));                      // 32 MB
    float* h = (float*)(ws + (64u << 10) + (32u << 20));        // 128 MB

    // 1) Orthonormalize hra_u
    hra_gs_kernel<<<1, 256, 0, stream>>>(hra, U);

    // 2) y = x Q = x - 2 (x U) U^T
    hra_apply_kernel<<<N_ROWS, 256, 0, stream>>>(x, U, y);

    // 3) h = gelu(y @ W_random)      M=4096, N=8192, K=2048
    dim3 g1(D_HID / BN, N_ROWS / BM);
    wmma_gemm_f32_kernel<0><<<g1, 256, 0, stream>>>(y, Wr, nullptr, h,
                                                    N_ROWS, D_HID, D_IN);

    // 4) out = h @ W_out + bias      M=4096, N=2048, K=8192
    dim3 g2(D_OUT / BN, N_ROWS / BM);
    wmma_gemm_f32_kernel<1><<<g2, 256, 0, stream>>>(h, Wo, bias, out,
                                                    N_ROWS, D_OUT, D_HID);
}